// CachedAttention_4861902979594
// MI455X (gfx1250) — compile-verified
//
#include <hip/hip_runtime.h>
#include <hip/hip_bf16.h>
#include <math.h>

typedef __attribute__((ext_vector_type(2))) float v2f;
typedef __attribute__((ext_vector_type(8))) float v8f;

#define DMODEL 1024
#define KVLEN  8192
#define NCHUNK 16
#define CHUNK  512     // KV rows per block
#define BATCH  16
#define KCH    32      // k-dims staged per async chunk (16 rows x 32 f32 = 2KB)

static __device__ __forceinline__ unsigned lds_addr_of(const void* p) {
  // generic (flat) shared pointer -> wave LDS byte offset (low 32 bits)
  return (unsigned)(uintptr_t)p;
}

// ---------------------------------------------------------------------------
// Kernel 1: q / new_k / new_v projections.  out = x @ W^T
//   M = 16 batches, N = 1024 outs, K = 1024.  One wave per 16x16 tile.
//   WMMA f32 16x16x4 (exact fp32 on the matrix pipe).
// ---------------------------------------------------------------------------
__global__ __launch_bounds__(32) void proj_kernel(
    const float* __restrict__ x,
    const float* __restrict__ wq,
    const float* __restrict__ wk,
    const float* __restrict__ wv,
    float* __restrict__ ws)          // ws: [q | nk | nv], each 16*1024 f32
{
  const int which = blockIdx.x >> 6;          // 0=q,1=k,2=v
  const int n0    = (blockIdx.x & 63) << 4;   // output-column tile base
  const float* W  = (which == 0) ? wq : (which == 1) ? wk : wv;
  float* out      = ws + (size_t)which * (BATCH * DMODEL);

  const int lane = threadIdx.x & 31;
  const int col  = lane & 15;
  const int h    = lane >> 4;

  const float* xrow = x + (size_t)col * DMODEL + 2 * h;      // A: row = batch
  const float* wrow = W + (size_t)(n0 + col) * DMODEL;       // B: col = out-dim

  v8f c = {};
  #pragma unroll 8
  for (int k0 = 0; k0 < DMODEL; k0 += 4) {
    v2f a = *(const v2f*)(xrow + k0);                        // x[col][k0+2h..+1]
    float4 wv4 = *(const float4*)(wrow + k0);                // W[n0+col][k0..k0+3]
    v2f b;
    b.x = h ? wv4.y : wv4.x;                                 // W[.][k0+h]
    b.y = h ? wv4.w : wv4.z;                                 // W[.][k0+2+h]
    c = __builtin_amdgcn_wmma_f32_16x16x4_f32(false, a, false, b,
                                              (short)0, c, false, false);
  }
  #pragma unroll
  for (int r = 0; r < 8; ++r) {
    const int m = r + 8 * h;
    out[(size_t)m * DMODEL + n0 + col] = c[r];
  }
}

// ---------------------------------------------------------------------------
// Kernel 2: split-KV flash-decode partial.
// grid = 16 batches * 16 chunks; block = 256 threads (8 waves).
// K tiles are staged into LDS with GLOBAL_LOAD_ASYNC_TO_LDS_B128 (double
// buffered, ASYNCcnt-pipelined), A-fragments come from ds_load_b64, scores
// via v_wmma_f32_16x16x4_f32 with q broadcast into all 16 B columns.
// PV accumulated in 32 regs/lane from coalesced float4 V loads.
// Deterministic cross-wave merge through LDS (no float atomics).
// ---------------------------------------------------------------------------
__global__ __launch_bounds__(256) void flash_partial_kernel(
    const float* __restrict__ cache_k,
    const float* __restrict__ cache_v,
    const float* __restrict__ qws,      // [16,1024]
    float* __restrict__ ml,             // [256][2]  (M, L) per block
    float* __restrict__ pacc)           // [256][1024]
{
  __shared__ float q_s[DMODEL];
  // 32KB region: during streaming = K stage buffers [8 waves][2 bufs][16*KCH];
  // after streaming (separated by __syncthreads) = accs[8][1024] merge buffer.
  __shared__ float smem[8 * 2 * 16 * KCH];
  __shared__ float red_m[8];
  __shared__ float red_l[8];

  const int b     = blockIdx.x / NCHUNK;
  const int chunk = blockIdx.x % NCHUNK;
  const int tid   = threadIdx.x;
  const int w     = tid >> 5;
  const int lane  = tid & 31;
  const int col   = lane & 15;
  const int h     = lane >> 4;

  for (int i = tid; i < DMODEL; i += 256) q_s[i] = qws[(size_t)b * DMODEL + i];
  __syncthreads();

  float* kst = &smem[w * (2 * 16 * KCH)];       // this wave's two K buffers
  const unsigned lds_buf0 = lds_addr_of(kst);
  const unsigned buf_bytes = 16 * KCH * 4;      // 2KB per buffer

  // staging lane map: async instr i moves rows 4i..4i+3 of the 16xKCH tile.
  const int srow = lane >> 3;                   // 0..3
  const int scol = (lane & 7) * 4;              // 0,4,...,28

  float m = -3.0e38f, l = 0.0f;
  float acc[32];
  #pragma unroll
  for (int j = 0; j < 32; ++j) acc[j] = 0.0f;

  const int row_base = chunk * CHUNK + w * 64;

  for (int t = 0; t < 4; ++t) {
    const int r0 = row_base + t * 16;
    const float* Kt = cache_k + ((size_t)b * KVLEN + r0) * DMODEL;

    // issue async stage of chunk kb into buffer `buf` (4 x b128 per lane-set)
    auto stage = [&](int kb, int buf) {
      const float*  g  = Kt + (size_t)srow * DMODEL + kb + scol;
      const unsigned ld = lds_buf0 + (unsigned)buf * buf_bytes +
                          (unsigned)(srow * KCH + scol) * 4;
      #pragma unroll
      for (int i = 0; i < 4; ++i) {
        asm volatile("global_load_async_to_lds_b128 %0, %1, off"
                     :: "v"(ld + (unsigned)(i * 4 * KCH * 4)),
                        "v"(g + (size_t)i * 4 * DMODEL)
                     : "memory");
      }
    };

    stage(0, 0);                                 // prefetch first chunk
    v8f c = {};
    for (int kb = 0; kb < DMODEL; kb += KCH) {
      const int cur = (kb / KCH) & 1;
      if (kb + KCH < DMODEL) {
        stage(kb + KCH, cur ^ 1);                // prefetch next chunk
        asm volatile("s_wait_asynccnt 0x4" ::: "memory");  // current resident
      } else {
        asm volatile("s_wait_asynccnt 0x0" ::: "memory");
      }
      const float* kc = kst + cur * (16 * KCH);
      #pragma unroll
      for (int k0 = 0; k0 < KCH; k0 += 4) {
        v2f a = *(const v2f*)&kc[col * KCH + k0 + 2 * h];   // ds_load_b64
        v2f bq;
        bq.x = q_s[kb + k0 + h];                            // B[h][*]
        bq.y = q_s[kb + k0 + 2 + h];                        // B[2+h][*]
        c = __builtin_amdgcn_wmma_f32_16x16x4_f32(false, a, false, bq,
                                                  (short)0, c, false, false);
      }
    }

    // all 16 scores per lane (columns of D are identical): swap halves.
    float s[16];
    #pragma unroll
    for (int r = 0; r < 8; ++r) {
      const float mine  = c[r];
      const float other = __shfl_xor(mine, 16, 32);
      s[r]     = h ? other : mine;          // rows 0..7
      s[r + 8] = h ? mine  : other;         // rows 8..15
    }

    const float scale = 0.03125f;           // 1/sqrt(1024)
    float tmax = -3.0e38f;
    #pragma unroll
    for (int r = 0; r < 16; ++r) { s[r] *= scale; tmax = fmaxf(tmax, s[r]); }

    const float mnew  = fmaxf(m, tmax);
    const float alpha = __expf(m - mnew);
    float p[16];
    float lsum = 0.0f;
    #pragma unroll
    for (int r = 0; r < 16; ++r) { p[r] = __expf(s[r] - mnew); lsum += p[r]; }
    l = l * alpha + lsum;
    m = mnew;
    #pragma unroll
    for (int j = 0; j < 32; ++j) acc[j] *= alpha;

    // PV: lane owns dims {j*128 + lane*4 .. +3}, coalesced float4 per row.
    const float* Vp = cache_v + ((size_t)b * KVLEN + r0) * DMODEL + lane * 4;
    for (int r = 0; r < 16; ++r) {
      const float pr = p[r];
      #pragma unroll
      for (int j = 0; j < 8; ++j) {
        float4 v = *(const float4*)(Vp + (size_t)r * DMODEL + j * 128);
        acc[j * 4 + 0] = fmaf(pr, v.x, acc[j * 4 + 0]);
        acc[j * 4 + 1] = fmaf(pr, v.y, acc[j * 4 + 1]);
        acc[j * 4 + 2] = fmaf(pr, v.z, acc[j * 4 + 2]);
        acc[j * 4 + 3] = fmaf(pr, v.w, acc[j * 4 + 3]);
      }
    }
  }

  // every wave drained its own ASYNCcnt above; after this barrier smem can be
  // safely repurposed as the cross-wave merge buffer accs[8][1024].
  if (lane == 0) { red_m[w] = m; red_l[w] = l; }
  __syncthreads();
  float (*accs)[DMODEL] = (float (*)[DMODEL])smem;

  float M = red_m[0];
  #pragma unroll
  for (int i = 1; i < 8; ++i) M = fmaxf(M, red_m[i]);

  const float myscale = __expf(m - M);
  #pragma unroll
  for (int j = 0; j < 8; ++j) {
    float4 vv;
    vv.x = acc[j * 4 + 0] * myscale;
    vv.y = acc[j * 4 + 1] * myscale;
    vv.z = acc[j * 4 + 2] * myscale;
    vv.w = acc[j * 4 + 3] * myscale;
    *(float4*)&accs[w][j * 128 + lane * 4] = vv;
  }
  __syncthreads();

  float L = 0.0f;
  #pragma unroll
  for (int i = 0; i < 8; ++i) L += red_l[i] * __expf(red_m[i] - M);

  if (tid == 0) {
    ml[(size_t)blockIdx.x * 2 + 0] = M;
    ml[(size_t)blockIdx.x * 2 + 1] = L;
  }
  // fixed-order wave sum (deterministic), thread t -> dims t*4..t*4+3
  const int d = tid * 4;
  float4 sum = make_float4(0.f, 0.f, 0.f, 0.f);
  #pragma unroll
  for (int i = 0; i < 8; ++i) {
    float4 vv = *(const float4*)&accs[i][d];
    sum.x += vv.x; sum.y += vv.y; sum.z += vv.z; sum.w += vv.w;
  }
  *(float4*)&pacc[(size_t)blockIdx.x * DMODEL + d] = sum;
}

// ---------------------------------------------------------------------------
// Kernel 3: combine 16 chunk-partials per batch + the new token (kv = 8192),
// normalize, round to 4 decimals (RNE, like jnp.round).
// ---------------------------------------------------------------------------
__global__ __launch_bounds__(256) void combine_kernel(
    const float* __restrict__ qws,
    const float* __restrict__ nkws,
    const float* __restrict__ nvws,
    const float* __restrict__ ml,
    const float* __restrict__ pacc,
    float* __restrict__ out)
{
  __shared__ float red[256];
  const int b   = blockIdx.x;
  const int tid = threadIdx.x;

  // q . new_k  (block reduction)
  const float* qb  = qws  + (size_t)b * DMODEL;
  const float* nkb = nkws + (size_t)b * DMODEL;
  float part = 0.0f;
  #pragma unroll
  for (int i = 0; i < 4; ++i) part += qb[tid * 4 + i] * nkb[tid * 4 + i];
  red[tid] = part;
  __syncthreads();
  for (int s2 = 128; s2 > 0; s2 >>= 1) {
    if (tid < s2) red[tid] += red[tid + s2];
    __syncthreads();
  }
  const float snew = red[0] * 0.03125f;

  float M = snew;
  #pragma unroll
  for (int i = 0; i < NCHUNK; ++i)
    M = fmaxf(M, ml[(size_t)(b * NCHUNK + i) * 2 + 0]);

  float L = __expf(snew - M);
  #pragma unroll
  for (int i = 0; i < NCHUNK; ++i) {
    const float mi = ml[(size_t)(b * NCHUNK + i) * 2 + 0];
    const float li = ml[(size_t)(b * NCHUNK + i) * 2 + 1];
    L += li * __expf(mi - M);
  }
  const float pnew = __expf(snew - M);

  const int d = tid * 4;
  float4 sum = make_float4(0.f, 0.f, 0.f, 0.f);
  #pragma unroll
  for (int i = 0; i < NCHUNK; ++i) {
    const float sc = __expf(ml[(size_t)(b * NCHUNK + i) * 2 + 0] - M);
    float4 a = *(const float4*)&pacc[(size_t)(b * NCHUNK + i) * DMODEL + d];
    sum.x = fmaf(a.x, sc, sum.x);
    sum.y = fmaf(a.y, sc, sum.y);
    sum.z = fmaf(a.z, sc, sum.z);
    sum.w = fmaf(a.w, sc, sum.w);
  }
  float4 nv = *(const float4*)(nvws + (size_t)b * DMODEL + d);
  const float invL = 1.0f / L;
  sum.x = (sum.x + pnew * nv.x) * invL;
  sum.y = (sum.y + pnew * nv.y) * invL;
  sum.z = (sum.z + pnew * nv.z) * invL;
  sum.w = (sum.w + pnew * nv.w) * invL;
  sum.x = nearbyintf(sum.x * 1e4f) * 1e-4f;
  sum.y = nearbyintf(sum.y * 1e4f) * 1e-4f;
  sum.z = nearbyintf(sum.z * 1e4f) * 1e-4f;
  sum.w = nearbyintf(sum.w * 1e4f) * 1e-4f;
  *(float4*)(out + (size_t)b * DMODEL + d) = sum;
}

// ---------------------------------------------------------------------------
extern "C" void kernel_launch(void* const* d_in, const int* in_sizes, int n_in,
                              void* d_out, int out_size, void* d_ws, size_t ws_size,
                              hipStream_t stream) {
  const float* x       = (const float*)d_in[0];   // [16,1,1024]
  const float* cache_k = (const float*)d_in[1];   // [16,8192,1024]
  const float* cache_v = (const float*)d_in[2];   // [16,8192,1024]
  const float* wq      = (const float*)d_in[3];   // [1024,1024]
  const float* wk      = (const float*)d_in[4];
  const float* wv      = (const float*)d_in[5];

  float* ws   = (float*)d_ws;
  float* q    = ws;                      // 16*1024
  float* nk   = ws + 1 * BATCH * DMODEL; // 16*1024
  float* nv   = ws + 2 * BATCH * DMODEL; // 16*1024
  float* ml   = ws + 3 * BATCH * DMODEL; // 256*2
  float* pacc = ml + 2 * BATCH * NCHUNK; // 256*1024  (16B-aligned)

  proj_kernel<<<3 * (DMODEL / 16), 32, 0, stream>>>(x, wq, wk, wv, ws);
  flash_partial_kernel<<<BATCH * NCHUNK, 256, 0, stream>>>(cache_k, cache_v, q, ml, pacc);
  combine_kernel<<<BATCH, 256, 0, stream>>>(q, nk, nv, ml, pacc, (float*)d_out);
}